// BilinearAttentionFusion2D_84954453115086
// MI455X (gfx1250) — compile-verified
//
#include <hip/hip_runtime.h>
#include <math.h>

typedef __attribute__((ext_vector_type(2))) float v2f;
typedef __attribute__((ext_vector_type(8))) float v8f;

#define B_SZ 512
#define DG   1024
#define P_SZ 1024
#define DA   256

// ---------- block-wide sum over 256 threads ----------
__device__ __forceinline__ float block_reduce_sum(float v, float* sm) {
  const int t = threadIdx.x;
  sm[t] = v;
  __syncthreads();
  for (int s = 128; s > 0; s >>= 1) {
    if (t < s) sm[t] += sm[t + s];
    __syncthreads();
  }
  float r = sm[0];
  __syncthreads();
  return r;
}

// ---------- fp32 WMMA GEMM:  C(MxN) = A(MxK,row) * Bm(NxK,row)^T ----------
// One wave -> 32(M) x 64(N) strip: 2 A-frags x 4 B-frags = 8 WMMAs per 6 loads.
// 4 waves/block stacked along M -> 128x64 tile.
// Distance-2 software pipeline with two parity-alternating buffers (no register
// rotation copies); '#pragma unroll 1' pins the hand schedule so the compiler
// cannot re-unroll and re-serialize it. Each buffer's loads have ~8 WMMAs + 6
// independent loads of slack before consumption.
// A frag (16x4): lane = m + 16*(k>>1), vgpr = k&1  -> per-lane float2 at A[m][k+2*kp]
// B frag (4x16): lane = n + 16*(k>>1), vgpr = k&1  -> per-lane float2 at Bm[n][k+2*kp]
// C/D (16x16):   lane = n + 16*(m>>3), vgpr = m&7
__global__ __launch_bounds__(128) void gemm_nt_wmma(
    const float* __restrict__ A, const float* __restrict__ Bm,
    float* __restrict__ C, int M, int N, int K) {
  const int wave = threadIdx.x >> 5;
  const int lane = threadIdx.x & 31;
  const int r    = lane & 15;
  const int kp   = lane >> 4;
  const int m0   = blockIdx.x * 128 + wave * 32;
  const int n0   = blockIdx.y * 64;

  const float* ap0 = A + (size_t)(m0 + r) * K + 2 * kp;
  const float* ap1 = ap0 + (size_t)16 * K;
  const float* bp0 = Bm + (size_t)(n0 + r) * K + 2 * kp;
  const float* bp1 = bp0 + (size_t)16 * K;
  const float* bp2 = bp1 + (size_t)16 * K;
  const float* bp3 = bp2 + (size_t)16 * K;

  v8f acc[2][4] = {};

  v2f aB0[2], bB0[4];   // buffer 0: k % 8 == 0 steps
  v2f aB1[2], bB1[4];   // buffer 1: k % 8 == 4 steps

  // prologue: fill both buffers (k = 0 and k = 4)
  aB0[0] = *(const v2f*)(ap0);     aB0[1] = *(const v2f*)(ap1);
  bB0[0] = *(const v2f*)(bp0);     bB0[1] = *(const v2f*)(bp1);
  bB0[2] = *(const v2f*)(bp2);     bB0[3] = *(const v2f*)(bp3);
  aB1[0] = *(const v2f*)(ap0 + 4); aB1[1] = *(const v2f*)(ap1 + 4);
  bB1[0] = *(const v2f*)(bp0 + 4); bB1[1] = *(const v2f*)(bp1 + 4);
  bB1[2] = *(const v2f*)(bp2 + 4); bB1[3] = *(const v2f*)(bp3 + 4);

#pragma unroll 1
  for (int k = 0; k < K - 8; k += 8) {
    // consume buffer 0 (step k), then refill it for step k+8
#pragma unroll
    for (int i = 0; i < 2; ++i)
#pragma unroll
      for (int j = 0; j < 4; ++j)
        acc[i][j] = __builtin_amdgcn_wmma_f32_16x16x4_f32(
            false, aB0[i], false, bB0[j], (short)0, acc[i][j], false, false);
    aB0[0] = *(const v2f*)(ap0 + k + 8); aB0[1] = *(const v2f*)(ap1 + k + 8);
    bB0[0] = *(const v2f*)(bp0 + k + 8); bB0[1] = *(const v2f*)(bp1 + k + 8);
    bB0[2] = *(const v2f*)(bp2 + k + 8); bB0[3] = *(const v2f*)(bp3 + k + 8);

    // consume buffer 1 (step k+4), then refill it for step k+12
#pragma unroll
    for (int i = 0; i < 2; ++i)
#pragma unroll
      for (int j = 0; j < 4; ++j)
        acc[i][j] = __builtin_amdgcn_wmma_f32_16x16x4_f32(
            false, aB1[i], false, bB1[j], (short)0, acc[i][j], false, false);
    aB1[0] = *(const v2f*)(ap0 + k + 12); aB1[1] = *(const v2f*)(ap1 + k + 12);
    bB1[0] = *(const v2f*)(bp0 + k + 12); bB1[1] = *(const v2f*)(bp1 + k + 12);
    bB1[2] = *(const v2f*)(bp2 + k + 12); bB1[3] = *(const v2f*)(bp3 + k + 12);
  }

  // epilogue: last two K steps (K-8 in buf0, K-4 in buf1)
#pragma unroll
  for (int i = 0; i < 2; ++i)
#pragma unroll
    for (int j = 0; j < 4; ++j)
      acc[i][j] = __builtin_amdgcn_wmma_f32_16x16x4_f32(
          false, aB0[i], false, bB0[j], (short)0, acc[i][j], false, false);
#pragma unroll
  for (int i = 0; i < 2; ++i)
#pragma unroll
    for (int j = 0; j < 4; ++j)
      acc[i][j] = __builtin_amdgcn_wmma_f32_16x16x4_f32(
          false, aB1[i], false, bB1[j], (short)0, acc[i][j], false, false);

#pragma unroll
  for (int i = 0; i < 2; ++i) {
    const int mrow = m0 + 16 * i + 8 * kp;
#pragma unroll
    for (int j = 0; j < 4; ++j) {
      const int nc = n0 + 16 * j + r;
#pragma unroll
      for (int v = 0; v < 8; ++v)
        C[(size_t)(mrow + v) * N + nc] = acc[i][j][v];
    }
  }
}

// ---------- stage A: wk/wv stats ----------
__global__ void stage_a_stats(const float* __restrict__ wk, const float* __restrict__ wv,
                              const float* __restrict__ gv,
                              float* wkc, float* wvc, float* wvg, float* scal) {
  __shared__ float sm[256];
  const int t = threadIdx.x;
  float wkt = wk[t];
  float mwk = block_reduce_sum(wkt, sm) * (1.0f / DA);
  float wc  = wkt - mwk;
  wkc[t] = wc;
  float vwk = block_reduce_sum(wc * wc, sm) * (1.0f / DA);
  float wvt = wv[t];
  float mwv = block_reduce_sum(wvt, sm) * (1.0f / DA);
  float wcv = wvt - mwv;
  wvc[t] = wcv;
  wvg[t] = wcv * gv[t];
  float vwv = block_reduce_sum(wcv * wcv, sm) * (1.0f / DA);
  if (t == 0) { scal[0] = vwk; scal[1] = vwv; }
}

// ---------- stage A2: Avec[e] = sum_d wkc*gk*Vk[e,d],  Bvec[e] = sum_d ln_k_b*Vk[e,d] ----------
__global__ void stage_a_vecs(const float* __restrict__ wkc, const float* __restrict__ gk,
                             const float* __restrict__ lnkb, const float* __restrict__ Vk,
                             float* Avec, float* Bvec) {
  __shared__ float wg[DA];
  __shared__ float bb[DA];
  const int t = threadIdx.x;
  wg[t] = wkc[t] * gk[t];
  bb[t] = lnkb[t];
  __syncthreads();
  float a = 0.f, b2 = 0.f;
  const float* row = Vk + (size_t)t * DA;
  for (int d = 0; d < DA; ++d) { float v = row[d]; a += wg[d] * v; b2 += bb[d] * v; }
  Avec[t] = a; Bvec[t] = b2;
}

// ---------- stage B: per-p centered embeddings + stats ----------
__global__ void stage_b(const float* __restrict__ bk, const float* __restrict__ Ek,
                        const float* __restrict__ gk,
                        const float* __restrict__ bv, const float* __restrict__ Ev,
                        const float* __restrict__ gv,
                        const float* __restrict__ wkc, const float* __restrict__ wvc,
                        float* varck, float* covk, float* cpckg,
                        float* varcv, float* covv, float* cvgT) {
  __shared__ float sm[256];
  const int p = blockIdx.x, t = threadIdx.x;

  float ck  = bk[t] + Ek[(size_t)p * DA + t];
  float mck = block_reduce_sum(ck, sm) * (1.0f / DA);
  float cc  = ck - mck;
  float vk_ = block_reduce_sum(cc * cc, sm) * (1.0f / DA);
  float cvk = block_reduce_sum(wkc[t] * cc, sm) * (1.0f / DA);
  if (t == 0) { varck[p] = vk_; covk[p] = cvk; }
  cpckg[(size_t)p * DA + t] = cc * gk[t];

  float cv  = bv[t] + Ev[(size_t)p * DA + t];
  float mcv = block_reduce_sum(cv, sm) * (1.0f / DA);
  float ccv = cv - mcv;
  float vv_ = block_reduce_sum(ccv * ccv, sm) * (1.0f / DA);
  float cvv = block_reduce_sum(wvc[t] * ccv, sm) * (1.0f / DA);
  if (t == 0) { varcv[p] = vv_; covv[p] = cvv; }
  cvgT[(size_t)t * P_SZ + p] = ccv * gv[t];   // store transposed (DA x P) for NT GEMM
}

// ---------- LayerNorm over rows of length DA (with pre-LN bias) ----------
__global__ void ln_rows(const float* __restrict__ x, const float* __restrict__ bias,
                        const float* __restrict__ g, const float* __restrict__ b,
                        float* __restrict__ y) {
  __shared__ float sm[256];
  const int row = blockIdx.x, t = threadIdx.x;
  float xv  = x[(size_t)row * DA + t] + bias[t];
  float mu  = block_reduce_sum(xv, sm) * (1.0f / DA);
  float d   = xv - mu;
  float var = block_reduce_sum(d * d, sm) * (1.0f / DA);
  y[(size_t)row * DA + t] = d * rsqrtf(var + 1e-5f) * g[t] + b[t];
}

// ---------- stage E: logits, sigmoid, normalize; emit alpha, w2, s0, s1 ----------
__global__ void stage_e(const float* __restrict__ desc, const float* __restrict__ q2,
                        const float* __restrict__ Avec, const float* __restrict__ Bvec,
                        const float* __restrict__ varck, const float* __restrict__ covk,
                        const float* __restrict__ varcv, const float* __restrict__ covv,
                        const float* __restrict__ scal, const float* __restrict__ Cq,
                        float* __restrict__ alpha_out, float* __restrict__ w2,
                        float* __restrict__ s0, float* __restrict__ s1) {
  __shared__ float sm[256];
  __shared__ float bc[3];
  const int b = blockIdx.x, t = threadIdx.x;
  const float vwk = scal[0], vwv = scal[1];

  float qv = q2[(size_t)b * DA + t];
  float pa = block_reduce_sum(qv * Avec[t], sm);
  float pb = block_reduce_sum(qv * Bvec[t], sm);
  if (t == 0) { bc[0] = pa; bc[1] = pb; }
  __syncthreads();
  const float a_b = bc[0], b_b = bc[1];

  float dv[4], au[4];
  float S = 0.f;
#pragma unroll
  for (int i = 0; i < 4; ++i) {
    const int p = t + i * 256;
    float d = desc[(size_t)b * P_SZ + p];
    float rk = rsqrtf(d * d * vwk + 2.0f * d * covk[p] + varck[p] + 1e-5f);
    float e  = (rk * (d * a_b + Cq[(size_t)b * P_SZ + p]) + b_b) * 0.0625f; // /sqrt(256)
    float a_un = 1.0f / (1.0f + expf(-e));
    dv[i] = d; au[i] = a_un; S += a_un;
  }
  S = block_reduce_sum(S, sm);
  if (t == 0) bc[2] = S;
  __syncthreads();
  const float Stot = bc[2];
  const float inv  = 1.0f / (Stot + 1e-12f);

  float s1p = 0.f;
#pragma unroll
  for (int i = 0; i < 4; ++i) {
    const int p = t + i * 256;
    float alpha = au[i] * inv;
    alpha_out[(size_t)b * P_SZ + p] = alpha;
    float d  = dv[i];
    float rv = rsqrtf(d * d * vwv + 2.0f * d * covv[p] + varcv[p] + 1e-5f);
    float w  = alpha * rv;
    w2[(size_t)b * P_SZ + p] = w;
    s1p += w * d;
  }
  s1p = block_reduce_sum(s1p, sm);
  if (t == 0) { s1[b] = s1p; s0[b] = Stot * inv; }
}

// ---------- epilogue: c += s1[b]*wvg[d] + s0[b]*ln_v_b[d] ----------
__global__ void epilogue_c(float* __restrict__ c, const float* __restrict__ s1,
                           const float* __restrict__ s0, const float* __restrict__ wvg,
                           const float* __restrict__ lnvb) {
  const int idx = blockIdx.x * 256 + threadIdx.x;
  const int b = idx >> 8, d = idx & 255;
  c[idx] += s1[b] * wvg[d] + s0[b] * lnvb[d];
}

extern "C" void kernel_launch(void* const* d_in, const int* in_sizes, int n_in,
                              void* d_out, int out_size, void* d_ws, size_t ws_size,
                              hipStream_t stream) {
  const float* hg   = (const float*)d_in[0];
  const float* desc = (const float*)d_in[1];
  const float* Wq   = (const float*)d_in[2];
  const float* bq   = (const float*)d_in[3];
  const float* wk   = (const float*)d_in[4];
  const float* bk   = (const float*)d_in[5];
  const float* wv   = (const float*)d_in[6];
  const float* bv   = (const float*)d_in[7];
  const float* Ek   = (const float*)d_in[8];
  const float* Ev   = (const float*)d_in[9];
  const float* Uq   = (const float*)d_in[10];
  const float* Vk   = (const float*)d_in[11];
  const float* lnqg = (const float*)d_in[12];
  const float* lnqb = (const float*)d_in[13];
  const float* lnkg = (const float*)d_in[14];
  const float* lnkb = (const float*)d_in[15];
  const float* lnvg = (const float*)d_in[16];
  const float* lnvb = (const float*)d_in[17];

  float* out = (float*)d_out;            // [0, B*DA): c  |  [B*DA, B*DA+B*P): alpha
  float* ws  = (float*)d_ws;

  float* wkc   = ws + 0;
  float* wvc   = ws + 256;
  float* wvg   = ws + 512;
  float* scal  = ws + 768;
  float* Avec  = ws + 1024;
  float* Bvec  = ws + 1280;
  float* varck = ws + 1536;
  float* covk  = ws + 2560;
  float* varcv = ws + 3584;
  float* covv  = ws + 4608;
  float* s0    = ws + 5632;
  float* s1    = ws + 6144;
  float* cpckg = ws + 8192;                  // P x DA
  float* cvgT  = cpckg + (size_t)P_SZ * DA;  // DA x P
  float* Cmat  = cvgT  + (size_t)DA * P_SZ;  // P x DA
  float* qpre  = Cmat  + (size_t)P_SZ * DA;  // B x DA
  float* qln   = qpre  + (size_t)B_SZ * DA;  // B x DA
  float* q2    = qln   + (size_t)B_SZ * DA;  // B x DA
  float* Cq    = q2    + (size_t)B_SZ * DA;  // B x P
  float* w2    = Cq    + (size_t)B_SZ * P_SZ;// B x P

  stage_a_stats<<<1, 256, 0, stream>>>(wk, wv, lnvg, wkc, wvc, wvg, scal);
  stage_a_vecs <<<1, 256, 0, stream>>>(wkc, lnkg, lnkb, Vk, Avec, Bvec);
  stage_b<<<P_SZ, 256, 0, stream>>>(bk, Ek, lnkg, bv, Ev, lnvg, wkc, wvc,
                                    varck, covk, cpckg, varcv, covv, cvgT);

  // Cmat = cpckg (PxDA) @ Vk(DAxDA)^T
  gemm_nt_wmma<<<dim3(P_SZ / 128, DA / 64), 128, 0, stream>>>(cpckg, Vk, Cmat, P_SZ, DA, DA);
  // qpre = hg (BxDG) @ Wq(DAxDG)^T
  gemm_nt_wmma<<<dim3(B_SZ / 128, DA / 64), 128, 0, stream>>>(hg, Wq, qpre, B_SZ, DA, DG);
  ln_rows<<<B_SZ, 256, 0, stream>>>(qpre, bq, lnqg, lnqb, qln);
  // q2 = qln @ Uq^T
  gemm_nt_wmma<<<dim3(B_SZ / 128, DA / 64), 128, 0, stream>>>(qln, Uq, q2, B_SZ, DA, DA);
  // Cq = q2 (BxDA) @ Cmat(PxDA)^T
  gemm_nt_wmma<<<dim3(B_SZ / 128, P_SZ / 64), 128, 0, stream>>>(q2, Cmat, Cq, B_SZ, P_SZ, DA);

  stage_e<<<B_SZ, 256, 0, stream>>>(desc, q2, Avec, Bvec, varck, covk, varcv, covv,
                                    scal, Cq, out + (size_t)B_SZ * DA, w2, s0, s1);

  // c_mid = w2 (BxP) @ cvgT(DAxP)^T
  gemm_nt_wmma<<<dim3(B_SZ / 128, DA / 64), 128, 0, stream>>>(w2, cvgT, out, B_SZ, DA, P_SZ);
  epilogue_c<<<(B_SZ * DA) / 256, 256, 0, stream>>>(out, s1, s0, wvg, lnvb);
}